// MBCGCN_TwoBehaviors_72026601554235
// MI455X (gfx1250) — compile-verified
//
#include <hip/hip_runtime.h>

#define EMBED 64

typedef __attribute__((ext_vector_type(2))) float v2f;
typedef __attribute__((ext_vector_type(8))) float v8f;

// Guaranteed-native fp32 atomic add (L2 RMW, no CAS loop, no return value).
__device__ __forceinline__ void gfadd(float* p, float v) {
  asm volatile("global_atomic_add_f32 %0, %1, off" : : "v"(p), "v"(v) : "memory");
}

// ---------------------------------------------------------------------------
// x0 = concat(user_emb, item_emb)
// ---------------------------------------------------------------------------
__global__ void k_init_x0(const float* __restrict__ u, const float* __restrict__ it,
                          float* __restrict__ x0,
                          long long nu_elems, long long tot_elems) {
  long long i = ((long long)blockIdx.x * blockDim.x + threadIdx.x) * 4;
  if (i >= tot_elems) return;
  float4 v;
  if (i < nu_elems) v = *(const float4*)(u + i);
  else              v = *(const float4*)(it + (i - nu_elems));
  *(float4*)(x0 + i) = v;
}

__global__ void k_zero(float* __restrict__ p, long long n) {
  long long i = ((long long)blockIdx.x * blockDim.x + threadIdx.x) * 4;
  if (i >= n) return;
  *(float4*)(p + i) = make_float4(0.f, 0.f, 0.f, 0.f);
}

// deg[col[e]] += 1
__global__ void k_degree(const int* __restrict__ col, float* __restrict__ deg, int ne) {
  int e = blockIdx.x * blockDim.x + threadIdx.x;
  if (e < ne) gfadd(deg + col[e], 1.0f);
}

// dinv = deg > 0 ? rsqrt(deg) : 0   (in place)
__global__ void k_rsqrt(float* __restrict__ d, int n) {
  int i = blockIdx.x * blockDim.x + threadIdx.x;
  if (i < n) {
    float v = d[i];
    d[i] = (v > 0.f) ? rsqrtf(v) : 0.f;
  }
}

// out[col] += x[row] * dinv[row]*dinv[col]   — one wave32 per edge, 2 floats/lane.
// 256B coalesced gather + 2 native f32 atomics per lane; x (38.4MB) fits in L2.
__global__ void k_scatter(const float* __restrict__ x, float* __restrict__ out,
                          const int* __restrict__ row, const int* __restrict__ col,
                          const float* __restrict__ dinv, int ne) {
  int lane = threadIdx.x & 31;
  int wpb  = blockDim.x >> 5;
  long long e = (long long)blockIdx.x * wpb + (threadIdx.x >> 5);
  if (e >= ne) return;                       // wave-uniform exit
  int r = row[e];
  int c = col[e];
  float w = dinv[r] * dinv[c];
  float2 v = *(const float2*)(x + (long long)r * EMBED + lane * 2);
  float* dst = out + (long long)c * EMBED + lane * 2;
  gfadd(dst,     v.x * w);
  gfadd(dst + 1, v.y * w);
}

// out = (a + b + c) / 3
__global__ void k_combine3(float* __restrict__ out, const float* __restrict__ a,
                           const float* __restrict__ b, const float* __restrict__ c,
                           long long n) {
  long long i = ((long long)blockIdx.x * blockDim.x + threadIdx.x) * 4;
  if (i >= n) return;
  const float s = 1.0f / 3.0f;
  float4 va = *(const float4*)(a + i);
  float4 vb = *(const float4*)(b + i);
  float4 vc = *(const float4*)(c + i);
  float4 o;
  o.x = (va.x + vb.x + vc.x) * s;
  o.y = (va.y + vb.y + vc.y) * s;
  o.z = (va.z + vb.z + vc.z) * s;
  o.w = (va.w + vb.w + vc.w) * s;
  *(float4*)(out + i) = o;
}

// out += (a + b + c) / 3
__global__ void k_combine3_add(float* __restrict__ out, const float* __restrict__ a,
                               const float* __restrict__ b, const float* __restrict__ c,
                               long long n) {
  long long i = ((long long)blockIdx.x * blockDim.x + threadIdx.x) * 4;
  if (i >= n) return;
  const float s = 1.0f / 3.0f;
  float4 va = *(const float4*)(a + i);
  float4 vb = *(const float4*)(b + i);
  float4 vc = *(const float4*)(c + i);
  float4 o  = *(float4*)(out + i);
  o.x += (va.x + vb.x + vc.x) * s;
  o.y += (va.y + vb.y + vc.y) * s;
  o.z += (va.z + vb.z + vc.z) * s;
  o.w += (va.w + vb.w + vc.w) * s;
  *(float4*)(out + i) = o;
}

// ---------------------------------------------------------------------------
// WMMA GEMM:  out[m,n] = sum_k x[m,k] * W_sel[n,k]   (W_sel = Wu if m<num_users)
// One wave per 16x16 tile; K=64 via 16 chained v_wmma_f32_16x16x4_f32.
// A frag (16x4 f32): lanes 0-15 hold M=lane, K={k0,k0+1}; lanes 16-31 K={k0+2,k0+3}.
// Since B[k][n] = W[n][k] row-major, the B frag is the same float2 row-major
// load pattern applied to W.
// ---------------------------------------------------------------------------
__global__ void k_gemm_wmma(const float* __restrict__ x,
                            const float* __restrict__ Wu,
                            const float* __restrict__ Wi,
                            float* __restrict__ out,
                            int num_users, int n_nodes) {
  int lane = threadIdx.x & 31;
  int wid  = (int)(((long long)blockIdx.x * blockDim.x + threadIdx.x) >> 5);
  int total_tiles = (n_nodes >> 4) * (EMBED / 16);
  if (wid >= total_tiles) return;            // wave-uniform, EXEC stays all-1s

  int m_base = (wid >> 2) << 4;
  int n_base = (wid & 3) << 4;
  const float* W = (m_base < num_users) ? Wu : Wi;

  int half = lane >> 4;       // K-half selector
  int l    = lane & 15;       // row within half
  const float* aptr = x + (long long)(m_base + l) * EMBED + 2 * half;
  const float* bptr = W + (n_base + l) * EMBED + 2 * half;

  v8f c = {};
#pragma unroll
  for (int k0 = 0; k0 < EMBED; k0 += 4) {
    v2f a = *(const v2f*)(aptr + k0);
    v2f b = *(const v2f*)(bptr + k0);
    c = __builtin_amdgcn_wmma_f32_16x16x4_f32(false, a, false, b, (short)0, c,
                                              false, false);
  }

  int n = n_base + l;
#pragma unroll
  for (int v = 0; v < 8; ++v) {
    int m = m_base + v + 8 * half;
    out[(long long)m * EMBED + n] = c[v];
  }
}

// ---------------------------------------------------------------------------
extern "C" void kernel_launch(void* const* d_in, const int* in_sizes, int n_in,
                              void* d_out, int out_size, void* d_ws, size_t ws_size,
                              hipStream_t stream) {
  const float* user_emb = (const float*)d_in[0];
  const float* item_emb = (const float*)d_in[1];
  const float* W_user   = (const float*)d_in[2];
  const float* W_item   = (const float*)d_in[3];
  const int*   e_cart   = (const int*)d_in[4];
  const int*   e_rent   = (const int*)d_in[5];

  const int num_users = in_sizes[0] / EMBED;   // 100000
  const int num_items = in_sizes[1] / EMBED;   // 50000
  const int n_nodes   = num_users + num_items; // 150000
  const int ne        = in_sizes[4] / 2;       // 2,000,000
  const long long NE  = (long long)n_nodes * EMBED;
  const long long NU  = (long long)num_users * EMBED;

  float* bufA = (float*)d_ws;    // x0 / rent0
  float* bufB = bufA + NE;       // h1
  float* bufC = bufB + NE;       // h2
  float* dinv = bufC + NE;       // degrees -> rsqrt, n_nodes floats
  float* outp = (float*)d_out;   // cart embeddings live here between phases

  const int T = 256;
  const int blkVec  = (int)((NE / 4 + T - 1) / T);              // float4 elementwise
  const int blkNode = (int)(((long long)n_nodes / 4 + T - 1) / T);
  const int blkDeg  = (ne + T - 1) / T;
  const int wpb     = T / 32;                                    // 8 edges per block
  const int blkEdge = (ne + wpb - 1) / wpb;
  const int blkNsc  = (n_nodes + T - 1) / T;
  const int total_tiles = (n_nodes / 16) * (EMBED / 16);
  const int blkGemm = (total_tiles + wpb - 1) / wpb;

  // ---------------- Phase A: cart behavior GCN ----------------
  k_init_x0<<<blkVec, T, 0, stream>>>(user_emb, item_emb, bufA, NU, NE);

  k_zero  <<<blkNode, T, 0, stream>>>(dinv, (long long)n_nodes);
  k_degree<<<blkDeg,  T, 0, stream>>>(e_cart + ne, dinv, ne);    // col = second half
  k_rsqrt <<<blkNsc,  T, 0, stream>>>(dinv, n_nodes);

  k_zero   <<<blkVec,  T, 0, stream>>>(bufB, NE);
  k_scatter<<<blkEdge, T, 0, stream>>>(bufA, bufB, e_cart, e_cart + ne, dinv, ne);
  k_zero   <<<blkVec,  T, 0, stream>>>(bufC, NE);
  k_scatter<<<blkEdge, T, 0, stream>>>(bufB, bufC, e_cart, e_cart + ne, dinv, ne);

  // cart = (x0 + h1 + h2) / 3  -> d_out
  k_combine3<<<blkVec, T, 0, stream>>>(outp, bufA, bufB, bufC, NE);

  // ---------------- Phase B: rent behavior GCN ----------------
  // rent0 = [cart_u @ Wu^T ; cart_i @ Wi^T]  -> bufA
  k_gemm_wmma<<<blkGemm, T, 0, stream>>>(outp, W_user, W_item, bufA,
                                         num_users, n_nodes);

  k_zero  <<<blkNode, T, 0, stream>>>(dinv, (long long)n_nodes);
  k_degree<<<blkDeg,  T, 0, stream>>>(e_rent + ne, dinv, ne);
  k_rsqrt <<<blkNsc,  T, 0, stream>>>(dinv, n_nodes);

  k_zero   <<<blkVec,  T, 0, stream>>>(bufB, NE);
  k_scatter<<<blkEdge, T, 0, stream>>>(bufA, bufB, e_rent, e_rent + ne, dinv, ne);
  k_zero   <<<blkVec,  T, 0, stream>>>(bufC, NE);
  k_scatter<<<blkEdge, T, 0, stream>>>(bufB, bufC, e_rent, e_rent + ne, dinv, ne);

  // out = cart + (rent0 + h1 + h2) / 3   (in place on d_out)
  k_combine3_add<<<blkVec, T, 0, stream>>>(outp, bufA, bufB, bufC, NE);
}